// NLP_embedder_68702296867162
// MI455X (gfx1250) — compile-verified
//
#include <hip/hip_runtime.h>

// Problem constants (from the reference)
#define LNUM   12
#define BNUM   16
#define HNUM   12
#define SNUM   256
#define VOCAB  30500
// rows = L*B*H*S = 589824, each row = 256 contiguous floats (k dim)

typedef float f4 __attribute__((ext_vector_type(4)));

// ---------------------------------------------------------------------------
// Zero the output histogram (L*H*VOCAB = 4,392,000 floats, divisible by 4).
// ---------------------------------------------------------------------------
__global__ __launch_bounds__(256) void zero_out_kernel(float* __restrict__ out, int n4) {
    int i = blockIdx.x * blockDim.x + threadIdx.x;
    if (i < n4) {
        f4 z = (f4){0.f, 0.f, 0.f, 0.f};
        *(f4*)(out + (size_t)i * 4) = z;
    }
}

// ---------------------------------------------------------------------------
// One wave per attention row (l,b,h,s). Staged early-exit:
//   stage 1: k in [0,32)    -> 1 float/lane  (128 B/row; passes w.p. 1 - 2^-32)
//   stage 2: k in [32,160)  -> float4/lane   (almost never executed)
//   stage 3: k in [160,256) -> float4, lanes 0..23
// If any element > 0.5, lane 0 atomically bumps out[(l*H+h)*VOCAB + ids[b,s]].
// ---------------------------------------------------------------------------
__global__ __launch_bounds__(256) void attn_hist_kernel(
    const float* __restrict__ attn,   // [L,B,H,S,S]
    const int*   __restrict__ ids,    // [B,S]
    float*       __restrict__ out,    // [L,H,VOCAB]
    int rows, int rowsPerWave)
{
    const int lane      = threadIdx.x & 31;
    const int waveInBlk = threadIdx.x >> 5;
    const int gwave     = blockIdx.x * 8 + waveInBlk;   // 8 waves per 256-thread block
    const int row0      = gwave * rowsPerWave;

    for (int r = 0; r < rowsPerWave; ++r) {
        const int row = row0 + r;
        if (row >= rows) return;
        const size_t base = (size_t)row * SNUM;

        // stage 1: first 32 keys, one cacheline-coalesced b32 load per wave
        float v = __builtin_nontemporal_load(attn + base + lane);
        unsigned m = __builtin_amdgcn_ballot_w32(v > 0.5f);

        if (m == 0u) {
            // stage 2: keys [32,160) — 128 floats as float4 per lane
            const f4* p2 = (const f4*)(attn + base + 32);
            f4 q = __builtin_nontemporal_load(p2 + lane);
            bool a2 = (q.x > 0.5f) || (q.y > 0.5f) || (q.z > 0.5f) || (q.w > 0.5f);
            m = __builtin_amdgcn_ballot_w32(a2);
            if (m == 0u) {
                // stage 3: keys [160,256) — 96 floats, lanes 0..23 read float4
                bool a3 = false;
                if (lane < 24) {
                    const f4* p3 = (const f4*)(attn + base + 160);
                    f4 q3 = __builtin_nontemporal_load(p3 + lane);
                    a3 = (q3.x > 0.5f) || (q3.y > 0.5f) || (q3.z > 0.5f) || (q3.w > 0.5f);
                }
                m = __builtin_amdgcn_ballot_w32(a3);
            }
        }

        if (m != 0u && lane == 0) {
            // row = ((l*B + b)*H + h)*S + s
            const int s  = row & (SNUM - 1);
            const int t  = row >> 8;        // (l*B + b)*H + h
            const int h  = t % HNUM;
            const int t2 = t / HNUM;        // l*B + b
            const int b  = t2 & (BNUM - 1);
            const int l  = t2 >> 4;
            const int id = ids[b * SNUM + s];
            atomicAdd(out + (size_t)(l * HNUM + h) * VOCAB + id, 1.0f);
        }
    }
}

// ---------------------------------------------------------------------------
// Host-side launcher
// ---------------------------------------------------------------------------
extern "C" void kernel_launch(void* const* d_in, const int* in_sizes, int n_in,
                              void* d_out, int out_size, void* d_ws, size_t ws_size,
                              hipStream_t stream) {
    const float* attn = (const float*)d_in[0];   // [L,B,H,S,S] float32
    const int*   ids  = (const int*)d_in[1];     // [B,S] integer
    float*       out  = (float*)d_out;           // [L,H,VOCAB] float32

    // 1) zero the histogram (harness poisons d_out; we must re-zero every call)
    const int n4 = out_size / 4;                 // 4,392,000 / 4 = 1,098,000
    zero_out_kernel<<<(n4 + 255) / 256, 256, 0, stream>>>(out, n4);

    // 2) staged threshold-reduce + scatter-add
    const int rows        = LNUM * BNUM * HNUM * SNUM;   // 589,824
    const int rowsPerWave = 8;
    const int wavesNeeded = (rows + rowsPerWave - 1) / rowsPerWave; // 73,728
    const int blocks      = (wavesNeeded + 7) / 8;                  // 9,216
    attn_hist_kernel<<<blocks, 256, 0, stream>>>(attn, ids, out, rows, rowsPerWave);
}